// DynamicRoIAlign_25804163514570
// MI455X (gfx1250) — compile-verified
//
#include <hip/hip_runtime.h>

// DynamicRoIAlign for MI455X (gfx1250).
// Memory-bound gather: feature map (105 MB) fits in the 192 MB L2, so corner
// gathers are L2 hits after first touch; output is streamed with non-temporal
// stores to keep the feature map resident. Per-ROI bilinear parameters are
// computed once into LDS (packed (offset,weight) int2 -> ds_load_b64) and
// reused across all 256 channels. Inner loop is 2-way unrolled for MLP.

#define C_CH   256
#define H_IN   160
#define W_IN   160
#define PLANE  (H_IN * W_IN)
#define SCALE_W 160.0f
#define SCALE_H 160.0f
#define THREADS 256
#define MAX_PIX 1024   // supports output grids up to 32x32; reference uses 14x14

__global__ __launch_bounds__(THREADS)
void roi_align_f32_kernel(const float* __restrict__ feats,
                          const float* __restrict__ rois,
                          const int*   __restrict__ pOH,
                          const int*   __restrict__ pOW,
                          float*       __restrict__ out)
{
    const int n  = blockIdx.x;
    const int OH = *pOH;
    const int OW = *pOW;
    const int nPix = OH * OW;
    const int tid  = threadIdx.x;

    // Per-pixel sampling parameters. Each corner k stores (plane offset,
    // validity-folded weight) packed in one int2 so phase 2 reads it with a
    // single ds_load_b64. Consecutive lanes -> consecutive p -> 8B stride
    // covers all 64 banks exactly once: conflict-free.
    __shared__ int2 s_par[4][MAX_PIX];

    // ROI scalars (uniform across block -> scalar loads).
    const float rb = rois[n * 5 + 0];
    const float x1 = rois[n * 5 + 1] * SCALE_W;
    const float y1 = rois[n * 5 + 2] * SCALE_H;
    const float x2 = rois[n * 5 + 3] * SCALE_W;
    const float y2 = rois[n * 5 + 4] * SCALE_H;
    const float rw = x2 - x1;
    const float rh = y2 - y1;

    // ---- Phase 1: bilinear parameters for each output pixel of this ROI ----
    for (int p = tid; p < nPix && p < MAX_PIX; p += THREADS) {
        const int oy = p / OW;
        const int ox = p - oy * OW;

        const float gx = (OW > 1) ? (float)ox / (float)(OW - 1) : 0.0f;
        const float gy = (OH > 1) ? (float)oy / (float)(OH - 1) : 0.0f;

        const float fx = x1 + gx * rw;
        const float fy = y1 + gy * rh;

        // ALIGNED == False path of the reference (grid_sample convention).
        const float nfx = fx / (float)W_IN * 2.0f - 1.0f;
        const float nfy = fy / (float)H_IN * 2.0f - 1.0f;
        const float ix  = ((nfx + 1.0f) * (float)W_IN - 1.0f) * 0.5f;
        const float iy  = ((nfy + 1.0f) * (float)H_IN - 1.0f) * 0.5f;

        const float x0f = floorf(ix);
        const float y0f = floorf(iy);
        const int   x0  = (int)x0f;
        const int   y0  = (int)y0f;
        const float wx1 = ix - x0f;
        const float wx0 = 1.0f - wx1;
        const float wy1 = iy - y0f;
        const float wy0 = 1.0f - wy1;

        #pragma unroll
        for (int k = 0; k < 4; ++k) {
            const int yi = y0 + (k >> 1);
            const int xi = x0 + (k & 1);
            const bool valid = (xi >= 0) && (xi < W_IN) && (yi >= 0) && (yi < H_IN);
            const int xc = min(max(xi, 0), W_IN - 1);
            const int yc = min(max(yi, 0), H_IN - 1);
            const float wgt = ((k >> 1) ? wy1 : wy0) * ((k & 1) ? wx1 : wx0);
            int2 pr;
            pr.x = yc * W_IN + xc;
            pr.y = __float_as_int(valid ? wgt : 0.0f);
            s_par[k][p] = pr;
        }
    }
    __syncthreads();

    // ---- Phase 2: stream all C*nPix outputs of this ROI -------------------
    const int b = (int)rb;
    const float* __restrict__ fplane0 = feats + (size_t)b * (size_t)(C_CH * PLANE);
    const int total = C_CH * nPix;
    float* __restrict__ out_n = out + (size_t)n * (size_t)total;
    const unsigned unPix = (unsigned)nPix;

    auto process = [&](int e) {
        const unsigned eu = (unsigned)e;
        const unsigned c  = eu / unPix;            // channel
        const unsigned p  = eu - c * unPix;        // pixel: consecutive across lanes

        const float* __restrict__ plane = fplane0 + (size_t)c * PLANE;

        const int2 pr0 = s_par[0][p];
        const int2 pr1 = s_par[1][p];
        const int2 pr2 = s_par[2][p];
        const int2 pr3 = s_par[3][p];

        const float v0 = plane[pr0.x];
        const float v1 = plane[pr1.x];
        const float v2 = plane[pr2.x];
        const float v3 = plane[pr3.x];

        float acc = __int_as_float(pr0.y) * v0;
        acc = fmaf(__int_as_float(pr1.y), v1, acc);
        acc = fmaf(__int_as_float(pr2.y), v2, acc);
        acc = fmaf(__int_as_float(pr3.y), v3, acc);

        // Write-once output: non-temporal so we don't evict the feature map
        // (which fits in and is served from the 192 MB L2).
        __builtin_nontemporal_store(acc, &out_n[e]);
    };

    int e = tid;
    // 2-way unrolled main loop: two independent gather groups in flight.
    for (; e + THREADS < total; e += 2 * THREADS) {
        process(e);
        process(e + THREADS);
    }
    for (; e < total; e += THREADS) {
        process(e);
    }
}

extern "C" void kernel_launch(void* const* d_in, const int* in_sizes, int n_in,
                              void* d_out, int out_size, void* d_ws, size_t ws_size,
                              hipStream_t stream)
{
    const float* feats = (const float*)d_in[0];
    const float* rois  = (const float*)d_in[1];
    const int*   pOH   = (const int*)d_in[2];
    const int*   pOW   = (const int*)d_in[3];
    float*       out   = (float*)d_out;

    const int N = in_sizes[1] / 5;   // number of ROIs

    roi_align_f32_kernel<<<N, THREADS, 0, stream>>>(feats, rois, pOH, pOW, out);
}